// SimpleGatedGCNLayer_43688407335306
// MI455X (gfx1250) — compile-verified
//
#include <hip/hip_runtime.h>

#define DFEAT 128
#define LDSW 132            // 128 + 4 pad dwords -> conflict-free column reads (132 % 64 = 4)
#define NSLOTS 256
#define BN_EPS 1e-5f

typedef float v2f __attribute__((ext_vector_type(2)));
typedef float v8f __attribute__((ext_vector_type(8)));
typedef unsigned int v4u __attribute__((ext_vector_type(4)));
typedef int v4i __attribute__((ext_vector_type(4)));
typedef int v8i __attribute__((ext_vector_type(8)));

__device__ __forceinline__ v8f wmma4(v2f a, v2f b, v8f c) {
  // V_WMMA_F32_16X16X4_F32 : D = A(16x4) * B(4x16) + C(16x16), fp32
  return __builtin_amdgcn_wmma_f32_16x16x4_f32(
      /*neg_a=*/false, a, /*neg_b=*/false, b,
      /*c_mod=*/(short)0, c, /*reuse_a=*/false, /*reuse_b=*/false);
}

// LDS byte offset of a __shared__ object (generic -> AS(3) addrspacecast)
__device__ __forceinline__ unsigned lds_off(const void* p) {
  return (unsigned)(unsigned long long)(__attribute__((address_space(3))) const char*)p;
}

// TDM: DMA a 16x128 fp32 tile (row stride 128 floats in memory) into LDS with
// 4-dword padding after each 128-dword row (matches LDSW=132 layout).
// D# per CDNA5 ISA; tracked on TENSORcnt.
__device__ __forceinline__ void tdm_load_tile_16x128(const float* gsrc,
                                                     unsigned lds_byte_off,
                                                     unsigned rows_total) {
  unsigned long long ga = (unsigned long long)(const void*)gsrc;
  v4u g0;
  g0[0] = 1u;                                            // count=1, no gather
  g0[1] = lds_byte_off;                                  // lds_addr
  g0[2] = (unsigned)ga;                                  // global_addr[31:0]
  g0[3] = ((unsigned)(ga >> 32) & 0x01FFFFFFu) | (2u << 30);  // addr[56:32] | type=2
  v8i g1;
  g1[0] = (int)((2u << 16)      // data_size = 4 bytes
              | (1u << 20)      // pad_enable
              | (6u << 22)      // pad_interval: 128 dwords
              | (3u << 25));    // pad_amount: 4 dwords
  g1[1] = (int)(128u << 16);                             // tensor_dim0 = 128
  g1[2] = (int)((rows_total & 0xFFFFu) << 16);           // tensor_dim1 lo
  g1[3] = (int)((rows_total >> 16) | (128u << 16));      // tensor_dim1 hi | tile_dim0=128
  g1[4] = 16;                                            // tile_dim1 = 16 rows
  g1[5] = 128;                                           // tensor_dim0_stride = 128
  g1[6] = 0;
  g1[7] = 0;
  v4i z4 = {0, 0, 0, 0};                                 // 2D tensor: groups 2/3 unused
  v8i z8 = {0, 0, 0, 0, 0, 0, 0, 0};
  // amdgpu-toolchain (clang-23) 6-arg form: (g0, g1, g2, g3, extra, cpol)
  __builtin_amdgcn_tensor_load_to_lds(g0, g1, z4, z4, z8, 0);
}

// prefetch one 512B feature row (4 cachelines) -> global_prefetch_b8
__device__ __forceinline__ void prefetch_row(const float* p) {
  __builtin_prefetch(p, 0, 0);
  __builtin_prefetch(p + 32, 0, 0);
  __builtin_prefetch(p + 64, 0, 0);
  __builtin_prefetch(p + 96, 0, 0);
}

// ---------------------------------------------------------------- zero fill
__global__ __launch_bounds__(256) void zero_kernel(float* __restrict__ p, size_t n) {
  size_t stride = (size_t)gridDim.x * blockDim.x;
  for (size_t i = (size_t)blockIdx.x * blockDim.x + threadIdx.x; i < n; i += stride)
    p[i] = 0.f;
}

// ---------------------------------------------------- Dh = h@Wd^T+bd, Eh = h@We^T+be
__global__ __launch_bounds__(256) void proj_nodes_kernel(
    const float* __restrict__ h,
    const float* __restrict__ Wd, const float* __restrict__ bd,
    const float* __restrict__ We, const float* __restrict__ be,
    float* __restrict__ Dh, float* __restrict__ Eh, int Nrows)
{
  __shared__ float Hs[16 * LDSW];
  const int tid  = threadIdx.x;
  const int lane = tid & 31, wave = tid >> 5;
  const size_t row0 = (size_t)blockIdx.x * 16;

  if (wave == 0) {
    tdm_load_tile_16x128(h + row0 * DFEAT, lds_off(Hs), (unsigned)Nrows);
    __builtin_amdgcn_s_wait_tensorcnt(0);
  }
  __syncthreads();

  const int col = wave * 16 + (lane & 15);
  const int kh  = (lane >> 4) * 2;      // K pair selector per half-wave
  const int mb  = (lane >> 4) * 8;      // C/D row base per half-wave
  v8f accd = {}; v8f acce = {};
  #pragma unroll 4
  for (int k = 0; k < DFEAT; k += 4) {
    v2f a  = *(const v2f*)&Hs[(lane & 15) * LDSW + k + kh];
    v2f b0 = *(const v2f*)(Wd + (size_t)col * DFEAT + k + kh);
    v2f b1 = *(const v2f*)(We + (size_t)col * DFEAT + k + kh);
    accd = wmma4(a, b0, accd);
    acce = wmma4(a, b1, acce);
  }
  const float bdv = bd[col], bev = be[col];
  #pragma unroll
  for (int r = 0; r < 8; ++r) {
    size_t row = row0 + mb + r;
    Dh[row * DFEAT + col] = accd[r] + bdv;
    Eh[row * DFEAT + col] = acce[r] + bev;
  }
}

// -------- per-edge: Ce=e@Wc^T+bc ; m=Dh[src]+Eh[dst]+Ce ; e_mlp=relu(m)@Wm^T+bm
__global__ __launch_bounds__(256) void edge_fused_kernel(
    const float* __restrict__ e,
    const float* __restrict__ Dh, const float* __restrict__ Eh,
    const float* __restrict__ Wc, const float* __restrict__ bc,
    const float* __restrict__ Wm, const float* __restrict__ bm,
    const int* __restrict__ src, const int* __restrict__ dst,
    float* __restrict__ e_mlp,
    float* __restrict__ slot_sum, float* __restrict__ slot_sq, int Erows)
{
  __shared__ float Es[16 * LDSW];
  __shared__ float Ms[16 * LDSW];
  __shared__ int   sS[16], sD[16];
  __shared__ float cstat[2 * DFEAT];

  const int tid  = threadIdx.x;
  const int lane = tid & 31, wave = tid >> 5;
  const size_t e0 = (size_t)blockIdx.x * 16;

  // wave0: TDM DMA of the e tile into LDS (padded rows)
  if (wave == 0) {
    tdm_load_tile_16x128(e + e0 * DFEAT, lds_off(Es), (unsigned)Erows);
  }
  // wave1: fetch edge endpoints and prefetch the gathered projection rows so
  // their HBM latency overlaps GEMM1
  if (wave == 1) {
    if (lane < 16) {
      int s = src[e0 + lane];
      sS[lane] = s;
      prefetch_row(Dh + (size_t)s * DFEAT);
    } else {
      int dn = dst[e0 + (lane - 16)];
      sD[lane - 16] = dn;
      prefetch_row(Eh + (size_t)dn * DFEAT);
    }
  }
  if (tid < 2 * DFEAT) cstat[tid] = 0.f;
  if (wave == 0) __builtin_amdgcn_s_wait_tensorcnt(0);
  __syncthreads();

  const int col = wave * 16 + (lane & 15);
  const int kh  = (lane >> 4) * 2;
  const int mb  = (lane >> 4) * 8;

  // GEMM1: Ce tile
  v8f acc = {};
  #pragma unroll 4
  for (int k = 0; k < DFEAT; k += 4) {
    v2f a = *(const v2f*)&Es[(lane & 15) * LDSW + k + kh];
    v2f b = *(const v2f*)(Wc + (size_t)col * DFEAT + k + kh);
    acc = wmma4(a, b, acc);
  }
  const float bcv = bc[col];
  #pragma unroll
  for (int r = 0; r < 8; ++r) {
    int row = mb + r;
    float v = acc[r] + bcv
            + Dh[(size_t)sS[row] * DFEAT + col]
            + Eh[(size_t)sD[row] * DFEAT + col];
    Ms[row * LDSW + col] = v > 0.f ? v : 0.f;   // ReLU -> LDS
  }
  __syncthreads();

  // GEMM2: edge MLP linear
  v8f acc2 = {};
  #pragma unroll 4
  for (int k = 0; k < DFEAT; k += 4) {
    v2f a = *(const v2f*)&Ms[(lane & 15) * LDSW + k + kh];
    v2f b = *(const v2f*)(Wm + (size_t)col * DFEAT + k + kh);
    acc2 = wmma4(a, b, acc2);
  }
  const float bmv = bm[col];
  float ps = 0.f, pq = 0.f;
  #pragma unroll
  for (int r = 0; r < 8; ++r) {
    int row = mb + r;
    float v = acc2[r] + bmv;
    e_mlp[(e0 + row) * DFEAT + col] = v;
    ps += v; pq += v * v;
  }
  // block reduce BN partials, then slot-hashed global accumulation
  atomicAdd(&cstat[col], ps);
  atomicAdd(&cstat[DFEAT + col], pq);
  __syncthreads();
  if (tid < DFEAT) {
    int slot = blockIdx.x & (NSLOTS - 1);
    atomicAdd(&slot_sum[slot * DFEAT + tid], cstat[tid]);
    atomicAdd(&slot_sq [slot * DFEAT + tid], cstat[DFEAT + tid]);
  }
}

// ------------------------------------------------ slots -> mean & rsqrt(var+eps)
__global__ void reduce_stats_kernel(const float* __restrict__ slot_sum,
                                    const float* __restrict__ slot_sq,
                                    float count,
                                    float* __restrict__ mu, float* __restrict__ rinv)
{
  int c = threadIdx.x;
  float s = 0.f, q = 0.f;
  for (int i = 0; i < NSLOTS; ++i) {
    s += slot_sum[i * DFEAT + c];
    q += slot_sq [i * DFEAT + c];
  }
  float m   = s / count;
  float var = q / count - m * m;   // biased variance
  mu[c]   = m;
  rinv[c] = rsqrtf(var + BN_EPS);
}

// ------------------- e_out = e + BN(e_mlp); scatter-add e_out into h_agg[dst]
// one float4 group per iteration
__global__ __launch_bounds__(256) void e_out_scatter_kernel(
    const float* __restrict__ e, const float* __restrict__ e_mlp,
    const float* __restrict__ gamma_e, const float* __restrict__ beta_e,
    const float* __restrict__ mu_e, const float* __restrict__ rinv_e,
    const int* __restrict__ dst,
    float* __restrict__ out_e, float* __restrict__ h_agg, size_t ngroups)
{
  size_t stride = (size_t)gridDim.x * blockDim.x;
  for (size_t g = (size_t)blockIdx.x * blockDim.x + threadIdx.x; g < ngroups;
       g += stride) {
    size_t edge = g >> 5;                 // 32 float4 groups per edge row
    int c = (int)(g & 31) * 4;
    float4 ev = ((const float4*)e)[g];
    float4 mv = ((const float4*)e_mlp)[g];
    float4 o;
    o.x = ev.x + gamma_e[c + 0] * (mv.x - mu_e[c + 0]) * rinv_e[c + 0] + beta_e[c + 0];
    o.y = ev.y + gamma_e[c + 1] * (mv.y - mu_e[c + 1]) * rinv_e[c + 1] + beta_e[c + 1];
    o.z = ev.z + gamma_e[c + 2] * (mv.z - mu_e[c + 2]) * rinv_e[c + 2] + beta_e[c + 2];
    o.w = ev.w + gamma_e[c + 3] * (mv.w - mu_e[c + 3]) * rinv_e[c + 3] + beta_e[c + 3];
    ((float4*)out_e)[g] = o;
    float* hb = h_agg + (size_t)dst[edge] * DFEAT + c;
    atomicAdd(hb + 0, o.x);               // global_atomic_add_f32
    atomicAdd(hb + 1, o.y);
    atomicAdd(hb + 2, o.z);
    atomicAdd(hb + 3, o.w);
  }
}

// ------------------------------------------------ per-feature stats of h_agg
__global__ __launch_bounds__(256) void h_stats_kernel(
    const float* __restrict__ h_agg,
    float* __restrict__ slot_sum, float* __restrict__ slot_sq, size_t total)
{
  __shared__ float cstat[2 * DFEAT];
  if (threadIdx.x < 2 * DFEAT) cstat[threadIdx.x] = 0.f;
  __syncthreads();
  size_t stride = (size_t)gridDim.x * blockDim.x;  // multiple of 128 -> fixed col/thread
  float ps = 0.f, pq = 0.f;
  for (size_t idx = (size_t)blockIdx.x * blockDim.x + threadIdx.x; idx < total;
       idx += stride) {
    float v = h_agg[idx];
    ps += v; pq += v * v;
  }
  int c = (int)(((size_t)blockIdx.x * blockDim.x + threadIdx.x) & (DFEAT - 1));
  atomicAdd(&cstat[c], ps);
  atomicAdd(&cstat[DFEAT + c], pq);
  __syncthreads();
  if (threadIdx.x < DFEAT) {
    int slot = blockIdx.x & (NSLOTS - 1);
    atomicAdd(&slot_sum[slot * DFEAT + threadIdx.x], cstat[threadIdx.x]);
    atomicAdd(&slot_sq [slot * DFEAT + threadIdx.x], cstat[DFEAT + threadIdx.x]);
  }
}

// ------------------------------------------------ h_out = h + BN(h_agg), float4
__global__ __launch_bounds__(256) void h_out_kernel(
    const float* __restrict__ h, const float* __restrict__ h_agg,
    const float* __restrict__ gamma_h, const float* __restrict__ beta_h,
    const float* __restrict__ mu_h, const float* __restrict__ rinv_h,
    float* __restrict__ out_h, size_t ngroups)
{
  size_t stride = (size_t)gridDim.x * blockDim.x;
  for (size_t g = (size_t)blockIdx.x * blockDim.x + threadIdx.x; g < ngroups;
       g += stride) {
    int c = (int)(g & 31) * 4;
    float4 hv = ((const float4*)h)[g];
    float4 av = ((const float4*)h_agg)[g];
    float4 o;
    o.x = hv.x + gamma_h[c + 0] * (av.x - mu_h[c + 0]) * rinv_h[c + 0] + beta_h[c + 0];
    o.y = hv.y + gamma_h[c + 1] * (av.y - mu_h[c + 1]) * rinv_h[c + 1] + beta_h[c + 1];
    o.z = hv.z + gamma_h[c + 2] * (av.z - mu_h[c + 2]) * rinv_h[c + 2] + beta_h[c + 2];
    o.w = hv.w + gamma_h[c + 3] * (av.w - mu_h[c + 3]) * rinv_h[c + 3] + beta_h[c + 3];
    ((float4*)out_h)[g] = o;
  }
}

extern "C" void kernel_launch(void* const* d_in, const int* in_sizes, int n_in,
                              void* d_out, int out_size, void* d_ws, size_t ws_size,
                              hipStream_t stream) {
  const float* h   = (const float*)d_in[0];
  const float* e   = (const float*)d_in[1];
  const float* u   = (const float*)d_in[2];
  const float* Wd  = (const float*)d_in[3];
  const float* bd  = (const float*)d_in[4];
  const float* We  = (const float*)d_in[5];
  const float* be  = (const float*)d_in[6];
  const float* Wc  = (const float*)d_in[7];
  const float* bc  = (const float*)d_in[8];
  const float* Wm  = (const float*)d_in[9];
  const float* bm  = (const float*)d_in[10];
  const float* gh  = (const float*)d_in[11];
  const float* bth = (const float*)d_in[12];
  const float* ge  = (const float*)d_in[13];
  const float* bte = (const float*)d_in[14];
  const int*   src = (const int*)d_in[15];
  const int*   dst = (const int*)d_in[16];
  // d_in[17] = revmap, unused by the forward computation

  const int N = in_sizes[0] / DFEAT;   // 32000
  const int E = in_sizes[1] / DFEAT;   // 512000

  // workspace layout (floats)
  float* ws       = (float*)d_ws;
  float* Dh       = ws;
  float* Eh       = Dh + (size_t)N * DFEAT;
  float* e_mlp    = Eh + (size_t)N * DFEAT;
  float* h_agg    = e_mlp + (size_t)E * DFEAT;
  float* slot_es  = h_agg + (size_t)N * DFEAT;
  float* slot_eq  = slot_es + (size_t)NSLOTS * DFEAT;
  float* slot_hs  = slot_eq + (size_t)NSLOTS * DFEAT;
  float* slot_hq  = slot_hs + (size_t)NSLOTS * DFEAT;
  float* mu_e     = slot_hq + (size_t)NSLOTS * DFEAT;
  float* rinv_e   = mu_e + DFEAT;
  float* mu_h     = rinv_e + DFEAT;
  float* rinv_h   = mu_h + DFEAT;

  float* out_h = (float*)d_out;
  float* out_e = out_h + (size_t)N * DFEAT;
  float* out_u = out_e + (size_t)E * DFEAT;

  // 1) zero h_agg + all stat slots (contiguous region)
  size_t zcount = (size_t)N * DFEAT + 4 * (size_t)NSLOTS * DFEAT + 4 * DFEAT;
  zero_kernel<<<4096, 256, 0, stream>>>(h_agg, zcount);

  // 2) node projections (WMMA, TDM-staged A tiles)
  proj_nodes_kernel<<<N / 16, 256, 0, stream>>>(h, Wd, bd, We, be, Dh, Eh, N);

  // 3) fused edge pipeline (2 WMMA GEMMs + prefetched gather + ReLU + BN partials)
  edge_fused_kernel<<<E / 16, 256, 0, stream>>>(e, Dh, Eh, Wc, bc, Wm, bm,
                                                src, dst, e_mlp, slot_es, slot_eq, E);

  // 4) edge BN stats
  reduce_stats_kernel<<<1, DFEAT, 0, stream>>>(slot_es, slot_eq, (float)E, mu_e, rinv_e);

  // 5) e_out + scatter-add into h_agg (float4 groups)
  size_t egroups = (size_t)E * (DFEAT / 4);
  e_out_scatter_kernel<<<32768, 256, 0, stream>>>(e, e_mlp, ge, bte, mu_e, rinv_e,
                                                  dst, out_e, h_agg, egroups);

  // 6) node BN stats
  size_t totN = (size_t)N * DFEAT;
  h_stats_kernel<<<2048, 256, 0, stream>>>(h_agg, slot_hs, slot_hq, totN);
  reduce_stats_kernel<<<1, DFEAT, 0, stream>>>(slot_hs, slot_hq, (float)N, mu_h, rinv_h);

  // 7) h_out (float4 groups)
  size_t hgroups = (size_t)N * (DFEAT / 4);
  h_out_kernel<<<4096, 256, 0, stream>>>(h, h_agg, gh, bth, mu_h, rinv_h, out_h, hgroups);

  // 8) u passthrough
  (void)hipMemcpyAsync(out_u, u, (size_t)E * DFEAT * sizeof(float),
                       hipMemcpyDeviceToDevice, stream);
}